// InfiniteMixturePrototype_79517024518219
// MI455X (gfx1250) — compile-verified
//
#include <hip/hip_runtime.h>
#include <math.h>
#include <stdint.h>

typedef float v2f __attribute__((ext_vector_type(2)));
typedef float v8f __attribute__((ext_vector_type(8)));

#define N_ROWS 8192
#define C_COLS 512
#define D_DIM  512
#define K_DIM  1024   // real||imag concatenated along K
#define BM     32     // rows per workgroup
#define KC     8      // K chunk staged in LDS (double-buffered)
#define NCH    (K_DIM / KC)
#define AP     12     // padded k-stride: 16B-aligned float4 groups, 16 distinct banks
#define SLP    516    // padded logits row stride

// LDS float offsets
#define SB_SZ  (C_COLS * AP)        // 6144 per buffer
#define SA_SZ  (BM * AP)            // 384 per buffer
#define SA_OFF (2 * SB_SZ)          // 12288
#define SMEM_F (2 * SB_SZ + 2 * SA_SZ)  // 13056 floats ~ 52 KB

// workspace layout in floats
#define WS_HSQ  0
#define WS_PSQ  8192
#define WS_PART 8704   // 256 blocks * 2 floats

// ---------- h_sq[n] = sum_d hr^2 + hi^2  (1024 contiguous floats per row) ----------
__global__ __launch_bounds__(256) void hsq_kernel(const float* __restrict__ h,
                                                  float* __restrict__ ws) {
  int wave = threadIdx.x >> 5, lane = threadIdx.x & 31;
  int row = blockIdx.x * 8 + wave;
  const float* p = h + (size_t)row * K_DIM;
  float s = 0.f;
  for (int d = lane; d < K_DIM; d += 32) { float v = p[d]; s += v * v; }
  for (int off = 16; off; off >>= 1) s += __shfl_xor(s, off, 32);
  if (lane == 0) ws[WS_HSQ + row] = s;
}

// ---------- p_sq[c] = sum_d pr^2 + pi^2 ----------
__global__ __launch_bounds__(256) void psq_kernel(const float* __restrict__ protos,
                                                  float* __restrict__ ws) {
  int wave = threadIdx.x >> 5, lane = threadIdx.x & 31;
  int c = blockIdx.x * 8 + wave;
  const float* pr = protos + (size_t)c * D_DIM;
  const float* pi = protos + (size_t)C_COLS * D_DIM + (size_t)c * D_DIM;
  float s = 0.f;
  for (int d = lane; d < D_DIM; d += 32) { float a = pr[d], b = pi[d]; s += a * a + b * b; }
  for (int off = 16; off; off >>= 1) s += __shfl_xor(s, off, 32);
  if (lane == 0) ws[WS_PSQ + c] = s;
}

__device__ __forceinline__ void async_b128_to_lds(const float* gptr, const float* lptr) {
  uint32_t lds_off = (uint32_t)(uintptr_t)lptr;          // addr[31:0] = LDS offset
  uint64_t gaddr   = (uint64_t)(uintptr_t)gptr;
  asm volatile("global_load_async_to_lds_b128 %0, %1, off"
               :: "v"(lds_off), "v"(gaddr) : "memory");
}

// stage one KC=8 chunk: every wave issues exactly 1 A-op + 4 B-ops (ASYNCcnt +5)
__device__ __forceinline__ void stage_chunk(const float* __restrict__ h,
                                            const float* __restrict__ protos,
                                            float* sA, float* sB,
                                            int row0, int kc0,
                                            int tid, int wave, int lane) {
  // A: 32 rows x 8 k = 64 float4 groups; lanes 0-7 of each wave
  if (lane < 8) {
    int g = wave * 8 + lane;          // 0..63
    int r = g >> 1, kk4 = (g & 1) * 4;
    async_b128_to_lds(h + (size_t)(row0 + r) * K_DIM + kc0 + kk4, &sA[r * AP + kk4]);
  }
  // B: 512 cols x 8 k = 1024 float4 groups; 4 per thread
  {
    int kk4   = (tid & 1) * 4;
    int cbase = tid >> 1;             // 0..127
    const float* src = protos +
        ((kc0 < D_DIM) ? (size_t)kc0 : ((size_t)C_COLS * D_DIM + (size_t)(kc0 - D_DIM)));
#pragma unroll
    for (int i = 0; i < 4; ++i) {
      int c = cbase + 128 * i;
      async_b128_to_lds(src + (size_t)c * D_DIM + kk4, &sB[c * AP + kk4]);
    }
  }
}

// ---------- fused WMMA GEMM (double-buffered async staging) + per-row epilogue ----------
__global__ __launch_bounds__(256) void logits_fused_kernel(
    const float* __restrict__ h, const float* __restrict__ protos,
    const float* __restrict__ radii, const int* __restrict__ y,
    const int* __restrict__ slabels, const int* __restrict__ nInit,
    const float* __restrict__ ws, float* __restrict__ partials) {
  __shared__ float smem[SMEM_F];
  __shared__ int   slab[C_COLS];
  __shared__ float rowLossW[BM], rowAcc[BM];

  const int tid  = threadIdx.x;
  const int wave = tid >> 5, lane = tid & 31;
  const int m = lane & 15, half = lane >> 4;
  const int row0 = blockIdx.x * BM;
  float* sL = smem;                   // logits overlay (16*516 = 8256 floats)

  for (int c = tid; c < C_COLS; c += 256) slab[c] = slabels[c];

  v8f acc[2][4];
#pragma unroll
  for (int rb = 0; rb < 2; ++rb)
#pragma unroll
    for (int t = 0; t < 4; ++t)
#pragma unroll
      for (int g = 0; g < 8; ++g) acc[rb][t][g] = 0.f;
  const int cb0 = wave * 64;  // each wave owns 4 contiguous 16-col tiles (both row blocks)

  // prologue: stage chunk 0 into buffer 0
  stage_chunk(h, protos, smem + SA_OFF, smem, row0, 0, tid, wave, lane);

  for (int i = 0; i < NCH; ++i) {
    const int cur = i & 1;
    float* sBc = smem + cur * SB_SZ;
    float* sAc = smem + SA_OFF + cur * SA_SZ;
    if (i + 1 < NCH) {
      // issue next chunk into the other buffer, then wait only for chunk i's 5 ops
      stage_chunk(h, protos, smem + SA_OFF + (cur ^ 1) * SA_SZ, smem + (cur ^ 1) * SB_SZ,
                  row0, (i + 1) * KC, tid, wave, lane);
      asm volatile("s_wait_asynccnt 0x5" ::: "memory");
    } else {
      asm volatile("s_wait_asynccnt 0x0" ::: "memory");
    }
    __syncthreads();   // chunk i visible block-wide

#pragma unroll
    for (int kk = 0; kk < KC; kk += 4) {
      // A frags: lanes 0-15 -> K=kk,kk+1 ; lanes 16-31 -> K=kk+2,kk+3
      v2f a0, a1;
      a0.x = sAc[m * AP + kk + 2 * half];
      a0.y = sAc[m * AP + kk + 2 * half + 1];
      a1.x = sAc[(16 + m) * AP + kk + 2 * half];
      a1.y = sAc[(16 + m) * AP + kk + 2 * half + 1];
#pragma unroll
      for (int t = 0; t < 4; ++t) {
        int col = cb0 + t * 16 + m;
        v2f b;
        b.x = sBc[col * AP + kk + 2 * half];
        b.y = sBc[col * AP + kk + 2 * half + 1];
        acc[0][t] = __builtin_amdgcn_wmma_f32_16x16x4_f32(
            false, a0, false, b, (short)0, acc[0][t], false, false);
        acc[1][t] = __builtin_amdgcn_wmma_f32_16x16x4_f32(
            false, a1, false, b, (short)0, acc[1][t], false, false);
      }
    }
    __syncthreads();   // reads done; buffer may be overwritten next iteration
  }

  const int nInitial = nInit[0];

  // two-phase epilogue: 16 rows at a time overlaid on the staging region
  for (int rb = 0; rb < 2; ++rb) {
#pragma unroll
    for (int t = 0; t < 4; ++t) {
      int col = cb0 + t * 16 + m;
      float inv2r = 0.5f / radii[col];
      float psq = ws[WS_PSQ + col];
#pragma unroll
      for (int g = 0; g < 8; ++g) {
        int row = g + 8 * half;  // C/D layout: VGPR g holds M=g / M=g+8
        float hsq = ws[WS_HSQ + row0 + rb * 16 + row];
        // logit = -(hsq + psq - 2*cross) / (2*radii)
        sL[row * SLP + col] = (2.0f * acc[rb][t][g] - hsq - psq) * inv2r;
      }
    }
    __syncthreads();

    for (int rr = 0; rr < 2; ++rr) {
      int row = wave * 2 + rr;               // 8 waves x 2 rows = 16
      const float* L = &sL[row * SLP];
      int n = row0 + rb * 16 + row;
      int yv = y[n];
      int label = (yv >= nInitial) ? -1 : yv;

      // pass 1: global argmax (first index) + restricted argmax over label class
      float mA = -INFINITY; int iA = 0;
      float mT = -INFINITY; int iT = 0;
      for (int c = lane; c < C_COLS; c += 32) {
        float v = L[c];
        if (v > mA) { mA = v; iA = c; }
        if (slab[c] == label && v > mT) { mT = v; iT = c; }
      }
      for (int off = 16; off; off >>= 1) {
        float ov = __shfl_xor(mA, off, 32); int oi = __shfl_xor(iA, off, 32);
        if (ov > mA || (ov == mA && oi < iA)) { mA = ov; iA = oi; }
        float tv = __shfl_xor(mT, off, 32); int ti = __shfl_xor(iT, off, 32);
        if (tv > mT || (tv == mT && ti < iT)) { mT = tv; iT = ti; }
      }
      int best = iT;                   // all -inf => 0, matches jnp.argmax
      int classBest = slab[best];
      float lBest = L[best];

      // pass 2: logsumexp denominator + per-class max for weight check
      float se = 0.f, m2 = -INFINITY;
      for (int c = lane; c < C_COLS; c += 32) {
        float v = L[c];
        se += expf(v - mA);
        if (slab[c] == classBest && v > m2) m2 = v;
      }
      for (int off = 16; off; off >>= 1) {
        se += __shfl_xor(se, off, 32);
        float o2 = __shfl_xor(m2, off, 32);
        if (o2 > m2) m2 = o2;
      }

      if (lane == 0) {
        float nll = (mA + logf(se)) - lBest;
        float w = (lBest == m2) ? 1.0f : 0.0f;
        rowLossW[rb * 16 + row] = nll * w;
        rowAcc[rb * 16 + row] = (slab[iA] == label) ? 1.0f : 0.0f;
      }
    }
    __syncthreads();  // protect sL before next phase overwrites / final sum
  }

  if (tid == 0) {
    float ls = 0.f, as = 0.f;
    for (int r = 0; r < BM; ++r) { ls += rowLossW[r]; as += rowAcc[r]; }
    partials[blockIdx.x * 2]     = ls;
    partials[blockIdx.x * 2 + 1] = as;
  }
}

// ---------- deterministic final reduction ----------
__global__ void finalize_kernel(const float* __restrict__ partials, float* __restrict__ out) {
  if (threadIdx.x == 0 && blockIdx.x == 0) {
    float ls = 0.f, as = 0.f;
    for (int i = 0; i < N_ROWS / BM; ++i) { ls += partials[2 * i]; as += partials[2 * i + 1]; }
    out[0] = ls / (float)N_ROWS;
    out[1] = as;
  }
}

extern "C" void kernel_launch(void* const* d_in, const int* in_sizes, int n_in,
                              void* d_out, int out_size, void* d_ws, size_t ws_size,
                              hipStream_t stream) {
  const float* h       = (const float*)d_in[0];
  const float* protos  = (const float*)d_in[1];
  const float* radii   = (const float*)d_in[2];
  const int*   y       = (const int*)d_in[3];
  const int*   slabels = (const int*)d_in[4];
  const int*   nInit   = (const int*)d_in[5];
  float* ws  = (float*)d_ws;
  float* out = (float*)d_out;

  hipLaunchKernelGGL(hsq_kernel, dim3(N_ROWS / 8), dim3(256), 0, stream, h, ws);
  hipLaunchKernelGGL(psq_kernel, dim3(C_COLS / 8), dim3(256), 0, stream, protos, ws);
  hipLaunchKernelGGL(logits_fused_kernel, dim3(N_ROWS / BM), dim3(256), 0, stream,
                     h, protos, radii, y, slabels, nInit, ws, ws + WS_PART);
  hipLaunchKernelGGL(finalize_kernel, dim3(1), dim3(32), 0, stream, ws + WS_PART, out);
}